// PackingEnergy_58256936403306
// MI455X (gfx1250) — compile-verified
//
#include <hip/hip_runtime.h>
#include <stdint.h>

// ---- problem constants (match reference setup) ----
namespace {
constexpr int BB   = 32;     // batch
constexpr int LL   = 4096;   // sequence length
constexpr int KK   = 64;     // neighbors
constexpr int NG   = 5;      // groups
constexpr int NAA  = 20;     // amino acids

constexpr int TILE  = 128;   // residues per block
constexpr int BLOCK = 128;   // threads per block (1 thread : 1 residue)

constexpr float MAXD      = 12.0f;  // max_dist
constexpr float RHALF     = 8.0f;   // BEST_R_HALF
constexpr float INV_TAU   = 5.0f;   // 1 / BEST_TAU
constexpr float VALID_EPS = 1e-4f;
}

__global__ __launch_bounds__(BLOCK) void hydro_packing_kernel(
    const int*   __restrict__ seq,    // [B,L]
    const float* __restrict__ r,      // [B,L,K]
    const int*   __restrict__ jidx,   // [B,L,K]
    const float* __restrict__ w_raw,  // [NAA]
    const float* __restrict__ nstar,  // [NAA,NG]
    const float* __restrict__ sigma,  // [NAA,NG]
    const int*   __restrict__ ga,     // [NAA] group assignment
    float*       __restrict__ outE,   // [B,L]
    float*       __restrict__ outN)   // [B,L,NG]
{
    __shared__ __align__(16) float s_r[TILE * KK];   // 32 KB staged distance tile
    __shared__ uint32_t s_grp32[LL / 4];             // 4 KB: group id (u8) per residue of row

    const int tid          = threadIdx.x;
    const int chunksPerRow = LL / TILE;              // 32
    const int b            = blockIdx.x / chunksPerRow;
    const int l0           = (blockIdx.x % chunksPerRow) * TILE;

    // ------------------------------------------------------------------
    // 1) Async global->LDS staging of the r tile (ASYNCcnt-tracked).
    //    GVS form: uniform tile base lives in an SGPR pair, lanes carry
    //    only a 32-bit byte offset -> no per-lane 64-bit address math.
    // ------------------------------------------------------------------
    {
        const uint64_t rbase =
            (uint64_t)(uintptr_t)(r + ((size_t)b * LL + (size_t)l0) * KK);
        const uint32_t lds0 = (uint32_t)(uintptr_t)(&s_r[0]); // LDS offset = addr[31:0]
        #pragma unroll
        for (int i = 0; i < (TILE * KK) / (BLOCK * 4); ++i) {   // 16 issues/thread
            const uint32_t byteOff = (uint32_t)(i * BLOCK + tid) * 16u;
            const uint32_t loff    = lds0 + byteOff;
            asm volatile("global_load_async_to_lds_b128 %0, %1, %2"
                         :: "v"(loff), "v"(byteOff), "s"(rbase)
                         : "memory");
        }
    }

    // ------------------------------------------------------------------
    // 2) While async copies fly, build group-of-residue row in LDS so the
    //    j_idx gather becomes an LDS byte gather.
    // ------------------------------------------------------------------
    {
        const int4* seq4 = (const int4*)(seq + (size_t)b * LL);
        #pragma unroll
        for (int i = 0; i < LL / 4 / BLOCK; ++i) {              // 8 iterations
            const int idx = i * BLOCK + tid;
            int4 sv = seq4[idx];
            uint32_t g0 = (uint32_t)ga[min(max(sv.x, 0), NAA - 1)] & 0xffu;
            uint32_t g1 = (uint32_t)ga[min(max(sv.y, 0), NAA - 1)] & 0xffu;
            uint32_t g2 = (uint32_t)ga[min(max(sv.z, 0), NAA - 1)] & 0xffu;
            uint32_t g3 = (uint32_t)ga[min(max(sv.w, 0), NAA - 1)] & 0xffu;
            s_grp32[idx] = g0 | (g1 << 8) | (g2 << 16) | (g3 << 24);
        }
    }

    // Drain this wave's async copies, then workgroup barrier (covers the
    // DS stores from step 2 and every other wave's async copies).
    asm volatile("s_wait_asynccnt 0" ::: "memory");
    __syncthreads();

    const unsigned char* s_grp = (const unsigned char*)s_grp32;

    // ------------------------------------------------------------------
    // 3) One thread per residue. 16-byte-chunk processing:
    //    - j_idx read as int4 (4x fewer vmem instructions, L1-friendly)
    //    - staged r read as ds_load_b128 with chunk rotation so the
    //      256 B/thread row stride stays (nearly) bank-conflict-free
    //    - sigmoid gate = v_exp + v_rcp
    // ------------------------------------------------------------------
    const int    l   = l0 + tid;
    const int4*  jg4 = (const int4*)(jidx + ((size_t)b * LL + (size_t)l) * KK);
    const int    rot = tid & 15;

    float acc0 = 0.f, acc1 = 0.f, acc2 = 0.f, acc3 = 0.f, acc4 = 0.f;

    #pragma unroll 4
    for (int c = 0; c < KK / 4; ++c) {                  // 16 chunks of 4
        const int   chunk = (c + rot) & 15;
        const int4  j4    = jg4[chunk];
        const float4 r4   = *(const float4*)(&s_r[tid * KK + chunk * 4]);

        const int   je[4] = {j4.x, j4.y, j4.z, j4.w};
        const float re[4] = {r4.x, r4.y, r4.z, r4.w};

        #pragma unroll
        for (int e = 0; e < 4; ++e) {
            const float rv = re[e];
            int jv = je[e];
            jv = min(max(jv, 0), LL - 1);
            const int g = (int)s_grp[jv];

            const float rm = fminf(rv, MAXD);
            const float ex = __expf((rm - RHALF) * INV_TAU);      // exp(-x)
            float gate = __builtin_amdgcn_rcpf(1.0f + ex);        // v_rcp_f32
            gate = (rv < MAXD - VALID_EPS) ? gate : 0.0f;

            acc0 += (g == 0) ? gate : 0.0f;
            acc1 += (g == 1) ? gate : 0.0f;
            acc2 += (g == 2) ? gate : 0.0f;
            acc3 += (g == 3) ? gate : 0.0f;
            acc4 += (g == 4) ? gate : 0.0f;
        }
    }

    // ------------------------------------------------------------------
    // 4) Per-residue Gaussian product epilogue (full precision, runs once).
    // ------------------------------------------------------------------
    const int s  = seq[(size_t)b * LL + (size_t)l];
    const int sc = min(max(s, 0), NAA - 1);

    const float wr = w_raw[sc];
    const float w  = (wr > 20.0f) ? wr : log1pf(expf(wr));   // softplus

    const float acc[NG] = {acc0, acc1, acc2, acc3, acc4};
    float lg = 0.0f;
    #pragma unroll
    for (int gq = 0; gq < NG; ++gq) {
        const float ns = nstar[sc * NG + gq];
        const float sg = sigma[sc * NG + gq];
        const float d  = acc[gq] - ns;
        lg -= (d * d) / (2.0f * sg * sg);
    }
    const float E = w * expf(lg);

    const size_t base = (size_t)b * LL + (size_t)l;
    outE[base] = E;
    float* on = outN + base * NG;
    on[0] = acc0; on[1] = acc1; on[2] = acc2; on[3] = acc3; on[4] = acc4;
}

extern "C" void kernel_launch(void* const* d_in, const int* in_sizes, int n_in,
                              void* d_out, int out_size, void* d_ws, size_t ws_size,
                              hipStream_t stream) {
    (void)in_sizes; (void)n_in; (void)out_size; (void)d_ws; (void)ws_size;

    const int*   seq   = (const int*)  d_in[0];   // [B,L] int32
    const float* r     = (const float*)d_in[1];   // [B,L,K] f32
    const int*   jidx  = (const int*)  d_in[2];   // [B,L,K] int32
    const float* w_raw = (const float*)d_in[3];   // [20] f32
    const float* nstar = (const float*)d_in[4];   // [20,5] f32
    const float* sigma = (const float*)d_in[5];   // [20,5] f32
    const int*   ga    = (const int*)  d_in[6];   // [20] int32

    float* outE = (float*)d_out;                  // [B,L]
    float* outN = outE + (size_t)BB * LL;         // [B,L,5]

    dim3 grid(BB * (LL / TILE));                  // 1024 blocks
    hydro_packing_kernel<<<grid, BLOCK, 0, stream>>>(seq, r, jidx, w_raw,
                                                     nstar, sigma, ga,
                                                     outE, outN);
}